// VoxelFeatureExtractor_45784351375623
// MI455X (gfx1250) — compile-verified
//
#include <hip/hip_runtime.h>

typedef __attribute__((ext_vector_type(2))) float v2f;
typedef __attribute__((ext_vector_type(8))) float v8f;

#define MPTS   32
#define CIN    4
#define COUT   64
#define NBLK1  256
#define BLK    256
#define NBLK2  640

// ---------------- Pass 1: moments of x over valid points -------------------
// Accumulate Sx[4], upper-tri Sxx[10], count -> per-block partials (16 floats/block)
__global__ void vfe_pass1(const float* __restrict__ x, const int* __restrict__ npt,
                          float* __restrict__ partials, int npoints) {
    float s0 = 0.f, s1 = 0.f, s2 = 0.f, s3 = 0.f;
    float q00 = 0.f, q01 = 0.f, q02 = 0.f, q03 = 0.f;
    float q11 = 0.f, q12 = 0.f, q13 = 0.f;
    float q22 = 0.f, q23 = 0.f, q33 = 0.f;
    float cnt = 0.f;

    int stride = gridDim.x * blockDim.x;
    for (int i = blockIdx.x * blockDim.x + threadIdx.x; i < npoints; i += stride) {
        int v = i >> 5;
        int p = i & 31;
        if (p < npt[v]) {
            const float4 xx = *(const float4*)(x + (size_t)i * 4);
            float x0 = xx.x, x1 = xx.y, x2 = xx.z, x3 = xx.w;
            s0 += x0; s1 += x1; s2 += x2; s3 += x3;
            q00 += x0 * x0; q01 += x0 * x1; q02 += x0 * x2; q03 += x0 * x3;
            q11 += x1 * x1; q12 += x1 * x2; q13 += x1 * x3;
            q22 += x2 * x2; q23 += x2 * x3; q33 += x3 * x3;
            cnt += 1.f;
        }
    }

    float vals[15] = { s0, s1, s2, s3, q00, q01, q02, q03, q11, q12, q13, q22, q23, q33, cnt };

    // wave32 shuffle reduction
    #pragma unroll
    for (int q = 0; q < 15; ++q) {
        float t = vals[q];
        #pragma unroll
        for (int off = 16; off >= 1; off >>= 1)
            t += __shfl_xor(t, off, 32);
        vals[q] = t;
    }

    __shared__ float lds[8][15];   // 256 threads -> 8 waves
    int lane = threadIdx.x & 31;
    int wave = threadIdx.x >> 5;
    if (lane == 0) {
        #pragma unroll
        for (int q = 0; q < 15; ++q) lds[wave][q] = vals[q];
    }
    __syncthreads();
    if (threadIdx.x < 15) {
        float t = 0.f;
        #pragma unroll
        for (int w = 0; w < 8; ++w) t += lds[w][threadIdx.x];
        partials[blockIdx.x * 16 + threadIdx.x] = t;
    }
}

// ------- Pass 1b: reduce partials, compute per-channel scale/shift ---------
__global__ void vfe_pass1b(const float* __restrict__ partials,
                           const float* __restrict__ W, const float* __restrict__ b,
                           const float* __restrict__ gamma, const float* __restrict__ beta,
                           float* __restrict__ sc, float* __restrict__ sh) {
    __shared__ float red[15];
    if (threadIdx.x < 15) {
        float s = 0.f;
        for (int i = 0; i < NBLK1; ++i) s += partials[i * 16 + threadIdx.x];
        red[threadIdx.x] = s;
    }
    __syncthreads();
    int c = threadIdx.x;
    if (c < COUT) {
        float Sx0 = red[0], Sx1 = red[1], Sx2 = red[2], Sx3 = red[3];
        float S00 = red[4], S01 = red[5], S02 = red[6], S03 = red[7];
        float S11 = red[8], S12 = red[9], S13 = red[10];
        float S22 = red[11], S23 = red[12], S33 = red[13];
        float cnt = red[14];
        float N = fmaxf(cnt, 1.0f);
        float w0 = W[c * 4 + 0], w1 = W[c * 4 + 1], w2 = W[c * 4 + 2], w3 = W[c * 4 + 3];
        float bc = b[c];
        float dot = w0 * Sx0 + w1 * Sx1 + w2 * Sx2 + w3 * Sx3;
        float quad = w0 * w0 * S00 + w1 * w1 * S11 + w2 * w2 * S22 + w3 * w3 * S33
                   + 2.f * (w0 * w1 * S01 + w0 * w2 * S02 + w0 * w3 * S03
                          + w1 * w2 * S12 + w1 * w3 * S13 + w2 * w3 * S23);
        float sumh  = dot + bc * cnt;                    // sum of h = Wx+b over valid
        float sumh2 = quad + 2.f * bc * dot + bc * bc * cnt;
        float mu  = sumh / N;
        float var = (sumh2 - 2.f * mu * sumh + mu * mu * cnt) / N;
        float s = gamma[c] * rsqrtf(var + 1e-5f);
        sc[c] = s;
        // WMMA in pass 2 produces W·x only, so fold bias here:
        // hn = (W·x + b - mu)*s + beta = (W·x)*s + (beta + (b - mu)*s)
        sh[c] = beta[c] + (bc - mu) * s;
    }
}

// ----- Pass 2: per-voxel WMMA GEMM + fused BN/ReLU/masked-mean -------------
// One wave per voxel iteration. A = 16x4 point tile, B = 4x16 W^T tile.
// Invalid points are zeroed in A; each then contributes exactly relu(sh_c),
// corrected once per channel: full = acc_all - (32-nv)*relu(sh_c).
__global__ void vfe_pass2(const float* __restrict__ x, const int* __restrict__ npt,
                          const float* __restrict__ W,
                          const float* __restrict__ sc, const float* __restrict__ sh,
                          float* __restrict__ out, int nvox) {
    const int lane = threadIdx.x & 31;
    const int lm = lane & 15;     // N index within tile / A row
    const int hb = lane >> 4;     // lane half -> K pair (A/B), M offset (D)

    const int wid = blockIdx.x * (blockDim.x >> 5) + (threadIdx.x >> 5);
    const int nw  = gridDim.x * (blockDim.x >> 5);

    // B tiles of W^T (4x16 each), ISA layout: lane half selects K pair, vgpr selects K within pair
    v2f   bt[4];
    float sct[4], sht[4], rsh[4];
    #pragma unroll
    for (int t = 0; t < 4; ++t) {
        int c = t * 16 + lm;
        bt[t]  = *(const v2f*)(W + c * 4 + hb * 2);
        sct[t] = sc[c];
        sht[t] = sh[c];
        rsh[t] = fmaxf(sht[t], 0.f);   // contribution of one zeroed (invalid) point
    }
    const v2f vz = {};
    const v8f czero = {};

    int v = wid;
    v2f alo = vz, ahi = vz;
    int nv = 0;
    if (v < nvox) {
        const float* xv = x + (size_t)v * (MPTS * CIN);
        alo = *(const v2f*)(xv + lm * 4 + hb * 2);          // points 0..15
        ahi = *(const v2f*)(xv + (16 + lm) * 4 + hb * 2);   // points 16..31
        nv  = npt[v];
    }

    while (v < nvox) {
        int vn = v + nw;
        v2f calo = alo, cahi = ahi;
        int cnv = nv;

        // software pipeline: prefetch next voxel while current computes
        if (vn < nvox) {
            const float* xv = x + (size_t)vn * (MPTS * CIN);
            alo = *(const v2f*)(xv + lm * 4 + hb * 2);
            ahi = *(const v2f*)(xv + (16 + lm) * 4 + hb * 2);
            nv  = npt[vn];
        }

        // zero invalid point rows (row index = lm / 16+lm, uniform across K in lane)
        v2f malo = (lm < cnv) ? calo : vz;
        v2f mahi = ((16 + lm) < cnv) ? cahi : vz;

        int inval = 32 - cnv; if (inval < 0) inval = 0;
        float kinv = (float)inval;
        float inv  = __builtin_amdgcn_rcpf((float)(cnv > 1 ? cnv : 1));

        float accs[4];
        #pragma unroll
        for (int t = 0; t < 4; ++t) {
            v8f d0 = __builtin_amdgcn_wmma_f32_16x16x4_f32(
                false, malo, false, bt[t], (short)0, czero, false, false);
            v8f d1 = __builtin_amdgcn_wmma_f32_16x16x4_f32(
                false, mahi, false, bt[t], (short)0, czero, false, false);

            float a = 0.f;
            #pragma unroll
            for (int r = 0; r < 8; ++r) {
                a += fmaxf(d0[r] * sct[t] + sht[t], 0.f);
                a += fmaxf(d1[r] * sct[t] + sht[t], 0.f);
            }
            // combine lane halves -> every lane holds full 32-point sum for channel t*16+lm
            a += __shfl_xor(a, 16, 32);
            accs[t] = (a - kinv * rsh[t]) * inv;
        }

        // full-width stores: lanes 0..15 carry tiles 0/2, lanes 16..31 tiles 1/3
        size_t o = (size_t)v * COUT;
        out[o + lane]      = (lane < 16) ? accs[0] : accs[1];   // channels 0..31
        out[o + 32 + lane] = (lane < 16) ? accs[2] : accs[3];   // channels 32..63

        v = vn;
    }
}

extern "C" void kernel_launch(void* const* d_in, const int* in_sizes, int n_in,
                              void* d_out, int out_size, void* d_ws, size_t ws_size,
                              hipStream_t stream) {
    const float* x     = (const float*)d_in[0];
    const int*   npt   = (const int*)d_in[1];
    const float* W     = (const float*)d_in[2];
    const float* b     = (const float*)d_in[3];
    const float* gamma = (const float*)d_in[4];
    const float* beta  = (const float*)d_in[5];
    float* out = (float*)d_out;
    float* ws  = (float*)d_ws;

    int nvox    = in_sizes[1];        // 40000
    int npoints = nvox * MPTS;

    float* partials = ws;                 // NBLK1*16 floats (all rewritten each call)
    float* sc       = ws + NBLK1 * 16;    // 64 floats
    float* sh       = sc + COUT;          // 64 floats

    vfe_pass1 <<<NBLK1, BLK, 0, stream>>>(x, npt, partials, npoints);
    vfe_pass1b<<<1, 64, 0, stream>>>(partials, W, b, gamma, beta, sc, sh);
    vfe_pass2 <<<NBLK2, BLK, 0, stream>>>(x, npt, W, sc, sh, out, nvox);
}